// GASPairCopula_11123965296696
// MI455X (gfx1250) — compile-verified
//
#include <hip/hip_runtime.h>
#include <hip/hip_bf16.h>

#define T_LEN 4096
#define EPS_F 1e-6f

typedef float v2f __attribute__((ext_vector_type(2)));
typedef float v8f __attribute__((ext_vector_type(8)));

__device__ __forceinline__ float softplus_f(float x) {
    // stable log(1+exp(x))
    return (x > 0.f) ? (x + log1pf(expf(-x))) : log1pf(expf(x));
}

// Continued-fraction for regularized incomplete beta (cephes/NR betacf).
__device__ __forceinline__ float betacf_f(float a, float b, float x) {
    const float FPMIN = 1e-30f;
    float qab = a + b, qap = a + 1.f, qam = a - 1.f;
    float cc = 1.f;
    float d = 1.f - qab * x / qap;
    if (fabsf(d) < FPMIN) d = FPMIN;
    d = 1.f / d;
    float h = d;
#pragma unroll 1
    for (int m = 1; m <= 64; ++m) {
        float fm = (float)m;
        float m2 = 2.f * fm;
        float aa = fm * (b - fm) * x / ((qam + m2) * (a + m2));
        d = 1.f + aa * d; if (fabsf(d) < FPMIN) d = FPMIN;
        cc = 1.f + aa / cc; if (fabsf(cc) < FPMIN) cc = FPMIN;
        d = 1.f / d;
        h *= d * cc;
        aa = -(a + fm) * (qab + fm) * x / ((a + m2) * (qap + m2));
        d = 1.f + aa * d; if (fabsf(d) < FPMIN) d = FPMIN;
        cc = 1.f + aa / cc; if (fabsf(cc) < FPMIN) cc = FPMIN;
        d = 1.f / d;
        float del = d * cc;
        h *= del;
        if (fabsf(del - 1.f) < 1e-7f) break;
    }
    return h;
}

__device__ __forceinline__ float betainc_reg(float a, float b, float x, float lbeta) {
    if (x <= 0.f) return 0.f;
    if (x >= 1.f) return 1.f;
    float bt = expf(a * logf(x) + b * logf(1.f - x) - lbeta);
    if (x < (a + 1.f) / (a + b + 2.f))
        return bt * betacf_f(a, b, x) / a;
    return 1.f - bt * betacf_f(b, a, 1.f - x) / b;
}

// ---------------- Kernel A: parallel inverse Student-t CDF -----------------
// x_t = stdtrit(clip(u_t), nu), y_t = stdtrit(clip(v_t), nu); 60-step bisection
// over betainc(nu/2, 0.5, z), identical recursion to the reference.
__global__ void invcdf_kernel(const float* __restrict__ u, const float* __restrict__ v,
                              const float* __restrict__ nu_un,
                              float* __restrict__ xs, float* __restrict__ ys) {
    int i = blockIdx.x * blockDim.x + threadIdx.x;
    if (i >= 2 * T_LEN) return;
    const float* src = (i < T_LEN) ? u : v;
    float* dst       = (i < T_LEN) ? xs : ys;
    int j            = (i < T_LEN) ? i : i - T_LEN;

    __builtin_prefetch(src + j, 0, 1);   // global_prefetch_b8 (probe-verified)

    float c = src[j];
    c = fminf(1.f - EPS_F, fmaxf(EPS_F, c));

    float nu = softplus_f(nu_un[0]) + 2.01f;
    float a = 0.5f * nu, b = 0.5f;
    float w = fminf(1.f, fmaxf(1e-12f, 2.f * fminf(c, 1.f - c)));
    float lbeta = lgammaf(a) + lgammaf(b) - lgammaf(a + b);

    float lo = 0.f, hi = 1.f;
#pragma unroll 1
    for (int it = 0; it < 60; ++it) {
        float mid = 0.5f * (lo + hi);
        bool right = betainc_reg(a, b, mid, lbeta) < w;
        lo = right ? mid : lo;
        hi = right ? hi : mid;
    }
    float z  = fmaxf(0.5f * (lo + hi), 1e-12f);
    float xq = sqrtf(nu * (1.f - z) / z);
    float sg = (c > 0.5f) ? 1.f : ((c < 0.5f) ? -1.f : 0.f);
    dst[j] = sg * xq;
}

// ---------------- Kernel B: sequential GAS recurrence ----------------------
// Stage x,y in LDS (block-cooperative), then one lane runs the scan; the
// score uses the closed-form d(ll)/d(theta) * (1 - theta^2), clipped to +-5.
__global__ void scan_kernel(const float* __restrict__ xs, const float* __restrict__ ys,
                            const float* __restrict__ omega_p, const float* __restrict__ A_p,
                            const float* __restrict__ B_p, const float* __restrict__ nu_un,
                            float* __restrict__ thetas) {
    __shared__ float sx[T_LEN];
    __shared__ float sy[T_LEN];
    for (int i = threadIdx.x; i < T_LEN; i += blockDim.x) {
        sx[i] = xs[i];
        sy[i] = ys[i];
    }
    __syncthreads();
    if (threadIdx.x == 0) {
        float omega = omega_p[0], Ac = A_p[0], Bc = B_p[0];
        float nu = softplus_f(nu_un[0]) + 2.01f;
        float f = omega / (1.f - Bc);
#pragma unroll 1
        for (int t = 0; t < T_LEN; ++t) {
            float th = tanhf(f);
            thetas[t] = th;
            float x = sx[t], y = sy[t];
            float r2 = th * th;
            float omr2 = 1.f - r2;
            float s2 = x * x + y * y;
            float p = x * y;
            float num = s2 - 2.f * th * p;
            float zeta = num / omr2;
            float dzeta = 2.f * (th * num - p * omr2) / (omr2 * omr2);
            float dll = th / omr2 - 0.5f * (nu + 2.f) * dzeta / (nu + zeta);
            float score = dll * omr2;                 // chain rule through tanh
            score = fminf(5.f, fmaxf(-5.f, score));
            f = omega + Ac * score + Bc * f;
        }
    }
}

// ---------------- Kernel C: log-lik + WMMA f32 reduction -------------------
// Each wave folds 64 ll values per v_wmma_f32_16x16x4_f32 (B = ones =>
// exact f32 row sums, accumulated in C). Column 0 of D lives in lanes 0 & 16.
__global__ void ll_reduce_kernel(const float* __restrict__ xs, const float* __restrict__ ys,
                                 const float* __restrict__ thetas, const float* __restrict__ nu_un,
                                 float* __restrict__ out0) {
    __shared__ float partials[64];
    const int lane = threadIdx.x & 31;
    const int wave = threadIdx.x >> 5;             // 128 threads -> 4 waves
    const int chunks_per_wave = (T_LEN / 64) / 4;  // 16

    float nu = softplus_f(nu_un[0]) + 2.01f;
    float lgc = lgammaf(0.5f * (nu + 2.f)) + lgammaf(0.5f * nu)
              - 2.f * lgammaf(0.5f * (nu + 1.f));

    v8f acc = {};
    v2f ones; ones.x = 1.f; ones.y = 1.f;

#pragma unroll 1
    for (int itc = 0; itc < chunks_per_wave; ++itc) {
        int base = (wave * chunks_per_wave + itc) * 64 + lane * 2;
        v2f a;
#pragma unroll
        for (int k = 0; k < 2; ++k) {
            int t = base + k;
            float x = xs[t], y = ys[t], th = thetas[t];
            float omr2 = 1.f - th * th;
            float zeta = (x * x + y * y - 2.f * th * x * y) / omr2;
            float ll = lgc - 0.5f * logf(omr2)
                     - 0.5f * (nu + 2.f) * logf(1.f + zeta / nu)
                     + 0.5f * (nu + 1.f) * (logf(1.f + x * x / nu) + logf(1.f + y * y / nu));
            if (k == 0) a.x = ll; else a.y = ll;
        }
        // D = A*ones + C : accumulates per-row sums, replicated across columns.
        acc = __builtin_amdgcn_wmma_f32_16x16x4_f32(false, a, false, ones,
                                                    (short)0, acc, false, false);
    }
    // C/D layout: lanes 0-15 -> N=lane, M=VGPR(0..7); lanes 16-31 -> M=8..15.
    // Column 0 (one copy of every row sum) = lane 0 VGPR0..7 + lane 16 VGPR0..7.
    if (lane == 0) {
#pragma unroll
        for (int r = 0; r < 8; ++r) partials[wave * 16 + r] = acc[r];
    }
    if (lane == 16) {
#pragma unroll
        for (int r = 0; r < 8; ++r) partials[wave * 16 + 8 + r] = acc[r];
    }
    __syncthreads();
    if (threadIdx.x == 0) {
        float s = 0.f;
#pragma unroll 1
        for (int i = 0; i < 64; ++i) s += partials[i];
        out0[0] = -s;
    }
}

extern "C" void kernel_launch(void* const* d_in, const int* in_sizes, int n_in,
                              void* d_out, int out_size, void* d_ws, size_t ws_size,
                              hipStream_t stream) {
    const float* u      = (const float*)d_in[0];
    const float* v      = (const float*)d_in[1];
    const float* omega  = (const float*)d_in[2];
    const float* A      = (const float*)d_in[3];
    const float* B      = (const float*)d_in[4];
    const float* nu_un  = (const float*)d_in[5];
    float* out = (float*)d_out;     // out[0] = -sum(ll), out[1..4096] = thetas
    float* xs = (float*)d_ws;       // T floats
    float* ys = xs + T_LEN;         // T floats
    (void)in_sizes; (void)n_in; (void)out_size; (void)ws_size;

    invcdf_kernel<<<(2 * T_LEN + 255) / 256, 256, 0, stream>>>(u, v, nu_un, xs, ys);
    scan_kernel<<<1, 256, 0, stream>>>(xs, ys, omega, A, B, nu_un, out + 1);
    ll_reduce_kernel<<<1, 128, 0, stream>>>(xs, ys, out + 1, nu_un, out);
}